// FlowExp_71554155151985
// MI455X (gfx1250) — compile-verified
//
#include <hip/hip_runtime.h>

// FlowExp (scaling-and-squaring of a velocity field) for MI455X / gfx1250.
// Memory/gather-bound: no matmul structure -> WMMA not applicable. Optimized
// for data movement: AoS float3 intermediates (one global_load_b96 per gather
// corner), explicit load-then-FMA phases for max memory-level parallelism,
// L2-resident ping-pong between d_ws and d_out, gather-row prefetches
// (global_prefetch_b8), and non-temporal final stores.

namespace {

constexpr int   kB     = 2;
constexpr int   kD     = 160;
constexpr int   kH     = 192;
constexpr int   kW     = 160;
constexpr int   kHW    = kH * kW;
constexpr int   kVox   = kD * kH * kW;     // 4,915,200 voxels per batch
constexpr int   kTotal = kB * kVox;        // 9,830,400 threads
constexpr float kInvScale = 1.0f / 256.0f; // 1 / 2^NSTEPS

// 12-byte packed vector so AoS accesses lower to single b96 loads/stores.
struct F3 { float z, y, x; };

__device__ __forceinline__ int wrapi(int a, int n) {
  int r = a % n;                 // n is a compile-time constant -> mulhi seq
  return r + ((r >> 31) & n);    // branchless fixup for negative a
}

// One squaring step: dst[p] = inb ? scale * sample(src, p + scale*src[p]) : 0
// SRC_SOA: src is (B,3,D,H,W) channel planes (the raw input, step 1).
// DST_SOA: dst is (B,3,D,H,W) channel planes (the final output, step 8).
// Otherwise buffers are AoS float3 per voxel (z,y,x contiguous).
template <bool SRC_SOA, bool DST_SOA>
__global__ __launch_bounds__(256) void flow_step(const float* __restrict__ src,
                                                 float* __restrict__ dst,
                                                 float scale) {
  const int tid = blockIdx.x * blockDim.x + threadIdx.x;
  if (tid >= kTotal) return;

  const int b   = (tid >= kVox) ? 1 : 0;
  const int lin = tid - b * kVox;
  const int x   = lin % kW;
  const int t   = lin / kW;
  const int y   = t % kH;
  const int z   = t / kH;

  // Per-batch base pointers (hoists the 64-bit plane-offset math).
  const float* __restrict__ sb = src + (size_t)3 * (size_t)b * (size_t)kVox;
  float* __restrict__       db = dst + (size_t)3 * (size_t)b * (size_t)kVox;
  const F3* __restrict__    s3 = reinterpret_cast<const F3*>(sb);
  F3* __restrict__          d3 = reinterpret_cast<F3*>(db);

  // ---- displacement at this voxel (coalesced) ---------------------------
  float vz, vy, vx;
  if constexpr (SRC_SOA) {
    vz = sb[lin];
    vy = sb[lin + kVox];
    vx = sb[lin + 2 * kVox];
  } else {
    const F3 v = s3[lin];                  // one b96 load
    vz = v.z; vy = v.y; vx = v.x;
  }
  vz *= scale; vy *= scale; vx *= scale;

  const float cz = (float)z + vz;
  const float cy = (float)y + vy;
  const float cx = (float)x + vx;

  const float fz = floorf(cz), fy = floorf(cy), fx = floorf(cx);
  const float wz1 = cz - fz, wy1 = cy - fy, wx1 = cx - fx;
  const float wz0 = 1.0f - wz1, wy0 = 1.0f - wy1, wx0 = 1.0f - wx1;

  const int iz0 = wrapi((int)fz, kD);
  const int iy0 = wrapi((int)fy, kH);
  const int ix0 = wrapi((int)fx, kW);
  const int iz1 = (iz0 + 1 == kD) ? 0 : iz0 + 1;
  const int iy1 = (iy0 + 1 == kH) ? 0 : iy0 + 1;
  const int ix1 = (ix0 + 1 == kW) ? 0 : ix0 + 1;

  const int r00 = iz0 * kHW + iy0 * kW;    // four (z,y) gather rows
  const int r01 = iz0 * kHW + iy1 * kW;
  const int r10 = iz1 * kHW + iy0 * kW;
  const int r11 = iz1 * kHW + iy1 * kW;

  const float w00 = wz0 * wy0, w01 = wz0 * wy1;
  const float w10 = wz1 * wy0, w11 = wz1 * wy1;
  const float w000 = w00 * wx0, w001 = w00 * wx1;
  const float w010 = w01 * wx0, w011 = w01 * wx1;
  const float w100 = w10 * wx0, w101 = w10 * wx1;
  const float w110 = w11 * wx0, w111 = w11 * wx1;

  float az, ay, ax;
  if constexpr (SRC_SOA) {
    // Step 1 only: gather from the three SoA channel planes.
    const float* p0 = sb;
    const float* p1 = sb + kVox;
    const float* p2 = sb + 2 * kVox;
    const int i000 = r00 + ix0, i001 = r00 + ix1;
    const int i010 = r01 + ix0, i011 = r01 + ix1;
    const int i100 = r10 + ix0, i101 = r10 + ix1;
    const int i110 = r11 + ix0, i111 = r11 + ix1;
    // Phase 1: issue all 24 loads.
    const float z000 = p0[i000], z001 = p0[i001], z010 = p0[i010], z011 = p0[i011];
    const float z100 = p0[i100], z101 = p0[i101], z110 = p0[i110], z111 = p0[i111];
    const float y000 = p1[i000], y001 = p1[i001], y010 = p1[i010], y011 = p1[i011];
    const float y100 = p1[i100], y101 = p1[i101], y110 = p1[i110], y111 = p1[i111];
    const float x000 = p2[i000], x001 = p2[i001], x010 = p2[i010], x011 = p2[i011];
    const float x100 = p2[i100], x101 = p2[i101], x110 = p2[i110], x111 = p2[i111];
    // Phase 2: weighted reduction.
    az = fmaf(w000, z000, fmaf(w001, z001, fmaf(w010, z010, fmaf(w011, z011,
         fmaf(w100, z100, fmaf(w101, z101, fmaf(w110, z110, w111 * z111)))))));
    ay = fmaf(w000, y000, fmaf(w001, y001, fmaf(w010, y010, fmaf(w011, y011,
         fmaf(w100, y100, fmaf(w101, y101, fmaf(w110, y110, w111 * y111)))))));
    ax = fmaf(w000, x000, fmaf(w001, x001, fmaf(w010, x010, fmaf(w011, x011,
         fmaf(w100, x100, fmaf(w101, x101, fmaf(w110, x110, w111 * x111)))))));
  } else {
    // Prefetch the four gather rows (global_prefetch_b8): overlaps the
    // weight computation with the L2/WGP$ access.
    __builtin_prefetch(s3 + (r00 + ix0), 0, 1);
    __builtin_prefetch(s3 + (r01 + ix0), 0, 1);
    __builtin_prefetch(s3 + (r10 + ix0), 0, 1);
    __builtin_prefetch(s3 + (r11 + ix0), 0, 1);
    // Phase 1: issue all 8 b96 corner loads (one clause, all in flight).
    const F3 c000 = s3[r00 + ix0];
    const F3 c001 = s3[r00 + ix1];
    const F3 c010 = s3[r01 + ix0];
    const F3 c011 = s3[r01 + ix1];
    const F3 c100 = s3[r10 + ix0];
    const F3 c101 = s3[r10 + ix1];
    const F3 c110 = s3[r11 + ix0];
    const F3 c111 = s3[r11 + ix1];
    // Phase 2: weighted reduction.
    az = fmaf(w000, c000.z, fmaf(w001, c001.z, fmaf(w010, c010.z, fmaf(w011, c011.z,
         fmaf(w100, c100.z, fmaf(w101, c101.z, fmaf(w110, c110.z, w111 * c111.z)))))));
    ay = fmaf(w000, c000.y, fmaf(w001, c001.y, fmaf(w010, c010.y, fmaf(w011, c011.y,
         fmaf(w100, c100.y, fmaf(w101, c101.y, fmaf(w110, c110.y, w111 * c111.y)))))));
    ax = fmaf(w000, c000.x, fmaf(w001, c001.x, fmaf(w010, c010.x, fmaf(w011, c011.x,
         fmaf(w100, c100.x, fmaf(w101, c101.x, fmaf(w110, c110.x, w111 * c111.x)))))));
  }

  // extrapolate=False: zero where the sample leaves [0, n-1] per axis.
  const bool inb = (cz >= 0.0f) && (cz <= (float)(kD - 1)) &&
                   (cy >= 0.0f) && (cy <= (float)(kH - 1)) &&
                   (cx >= 0.0f) && (cx <= (float)(kW - 1));
  const float os = inb ? scale : 0.0f;
  az *= os; ay *= os; ax *= os;

  if constexpr (DST_SOA) {
    // Final write-once output: non-temporal (TH=NT), keep L2 for gathers.
    __builtin_nontemporal_store(az, &db[lin]);
    __builtin_nontemporal_store(ay, &db[lin + kVox]);
    __builtin_nontemporal_store(ax, &db[lin + 2 * kVox]);
  } else {
    F3 o; o.z = az; o.y = ay; o.x = ax;
    d3[lin] = o;                           // one b96 store
  }
}

}  // namespace

extern "C" void kernel_launch(void* const* d_in, const int* in_sizes, int n_in,
                              void* d_out, int out_size, void* d_ws, size_t ws_size,
                              hipStream_t stream) {
  (void)in_sizes; (void)n_in; (void)out_size; (void)ws_size;
  const float* flow = (const float*)d_in[0];  // (B,3,D,H,W) fp32
  float* out = (float*)d_out;                 // (B,3,D,H,W) fp32
  float* ws  = (float*)d_ws;                  // AoS float3 ping buffer,
                                              // kTotal*3 floats == out_size

  const int threads = 256;                    // 8 waves (wave32) per block
  const int blocks  = (kTotal + threads - 1) / threads;

  // 8 squaring steps; ping-pong AoS buffers between d_ws and d_out so only
  // one 118 MB scratch buffer is needed. Steps where dst==d_out hold AoS
  // intermediates except the final step, which writes the SoA result.
  flow_step<true,  false><<<blocks, threads, 0, stream>>>(flow, ws,  kInvScale);
  flow_step<false, false><<<blocks, threads, 0, stream>>>(ws,  out, 1.0f);
  flow_step<false, false><<<blocks, threads, 0, stream>>>(out, ws,  1.0f);
  flow_step<false, false><<<blocks, threads, 0, stream>>>(ws,  out, 1.0f);
  flow_step<false, false><<<blocks, threads, 0, stream>>>(out, ws,  1.0f);
  flow_step<false, false><<<blocks, threads, 0, stream>>>(ws,  out, 1.0f);
  flow_step<false, false><<<blocks, threads, 0, stream>>>(out, ws,  1.0f);
  flow_step<false, true ><<<blocks, threads, 0, stream>>>(ws,  out, 1.0f);
}